// EquivariantProductBasisBlock_57440892617129
// MI455X (gfx1250) — compile-verified
//
#include <hip/hip_runtime.h>
#include <hip/hip_bf16.h>

// ---------------------------------------------------------------------------
// MACE EquivariantProductBasisBlock for MI455X (gfx1250, wave32)
//
// Phase 1: per-element fused basis tensors  B3/B2/B1  (U (.) W over k)
// Phase 2: per-node symmetric contraction (Horner chain over i,j,n)
// Phase 3: o3.Linear channel mix as 9 GEMMs [4096x128]@[128x128] using
//          full-precision V_WMMA_F32_16X16X4_F32, + norm + sc, scattered
//          into the concatenated irrep layout.
// ---------------------------------------------------------------------------

typedef float v2f __attribute__((ext_vector_type(2)));
typedef float v8f __attribute__((ext_vector_type(8)));

#define NNODES 4096
#define NCH    128
#define NE     10
#define NIRR   9      // spherical components of node_feats
#define NM     9      // total output components (1 + 3 + 5)
#define OUTW   1152   // C * (1 + 3 + 5)

// workspace float counts
#define SZ_B3  (10L * 9 * 9 * 9 * 9 * 128)   // [e][m][i][j][n][c] = 8,398,080
#define SZ_B2  (10L * 9 * 9 * 9 * 128)       // [e][m][i][j][c]    =   933,120
#define SZ_B1  (10L * 9 * 9 * 128)           // [e][m][i][c]       =   103,680
#define SZ_RES (9L * 4096 * 128)             // [m][b][c]          = 4,718,592

// ---------------------------------------------------------------------------
// Phase 1: B3[e,m,i,j,n,c] = sum_k U3[d,i,j,n,k] W3[e,k,c]   (K3 = 30)
//          B2[e,m,i,j,c]   = sum_k U2[d,i,j,k]   W2[e,k,c]   (K2 = 12)
//          B1[e,m,i,c]     = sum_k U1[d,i,k]     W1[e,k,c]   (K1 = 4)
// grid: (dcount*81, 10)  block: 128 (c)
// ---------------------------------------------------------------------------
__global__ void precompute_kernel(const float* __restrict__ U3,
                                  const float* __restrict__ U2,
                                  const float* __restrict__ U1,
                                  const float* __restrict__ W3,
                                  const float* __restrict__ W2,
                                  const float* __restrict__ W1,
                                  float* __restrict__ B3,
                                  float* __restrict__ B2,
                                  float* __restrict__ B1,
                                  int mbase) {
  const int e   = blockIdx.y;
  const int idx = blockIdx.x;          // d*81 + i*9 + j
  const int d   = idx / 81;
  const int ij  = idx % 81;
  const int i   = ij / 9;
  const int j   = ij % 9;
  const int c   = threadIdx.x;         // channel
  const int m   = mbase + d;

  // hoist element-dependent channel weights (uniform over n)
  float w[30];
#pragma unroll
  for (int k = 0; k < 30; ++k) w[k] = W3[(e * 30 + k) * NCH + c];

#pragma unroll
  for (int n = 0; n < 9; ++n) {
    const float* u = &U3[(((long)(d * 9 + i) * 9 + j) * 9 + n) * 30];
    float s = 0.f;
#pragma unroll
    for (int k = 0; k < 30; ++k) s += u[k] * w[k];
    B3[((((((long)e * 9 + m) * 9 + i) * 9 + j) * 9 + n)) * NCH + c] = s;
  }

  {
    const float* u = &U2[((long)(d * 9 + i) * 9 + j) * 12];
    float s = 0.f;
#pragma unroll
    for (int k = 0; k < 12; ++k) s += u[k] * W2[(e * 12 + k) * NCH + c];
    B2[(((((long)e * 9 + m) * 9 + i) * 9 + j)) * NCH + c] = s;
  }

  if (j == 0) {
    const float* u = &U1[(long)(d * 9 + i) * 4];
    float s = 0.f;
#pragma unroll
    for (int k = 0; k < 4; ++k) s += u[k] * W1[(e * 4 + k) * NCH + c];
    B1[((((long)e * 9 + m) * 9 + i)) * NCH + c] = s;
  }
}

// ---------------------------------------------------------------------------
// Phase 2: per-node contraction.
// One wave (32 lanes) per node, each lane owns 4 consecutive channels so every
// B3/B2/B1 access is a coalesced float4 (global_load_b128).  B3 (33.6 MB) is
// L2-resident; ~3.4 MB of L2 reads per node.
//   res[m][b][c] = sum_i x_i * (B1 + sum_j x_j * (B2 + sum_n B3 * x_n))
// ---------------------------------------------------------------------------
__global__ void node_contract_kernel(const float* __restrict__ nf,
                                     const float* __restrict__ attrs,
                                     const float* __restrict__ B3,
                                     const float* __restrict__ B2,
                                     const float* __restrict__ B1,
                                     float* __restrict__ res) {
  __shared__ float xs[9 * NCH];
  __shared__ int zsh;

  const int b  = blockIdx.x;
  const int t  = threadIdx.x;   // 0..31
  const int c4 = t * 4;

  if (t == 0) {
    int z = 0;
    for (int e = 0; e < NE; ++e)
      if (attrs[b * NE + e] > 0.5f) z = e;
    zsh = z;
  }

  float xr[4][9];
#pragma unroll
  for (int ch = 0; ch < 4; ++ch)
#pragma unroll
    for (int n = 0; n < 9; ++n) {
      float v = nf[((long)b * NCH + c4 + ch) * NIRR + n];
      xr[ch][n] = v;
      xs[n * NCH + c4 + ch] = v;
    }
  __syncthreads();

  const int e = zsh;
  const long base3 = (long)e * (9L * 9 * 9 * 9 * NCH);
  const long base2 = (long)e * (9L * 9 * 9 * NCH);
  const long base1 = (long)e * (9L * 9 * NCH);

  for (int m = 0; m < NM; ++m) {
    float r0 = 0.f, r1 = 0.f, r2 = 0.f, r3 = 0.f;
    for (int i = 0; i < 9; ++i) {
      float4 xi = *(const float4*)&xs[i * NCH + c4];
      float4 a  = *(const float4*)&B1[base1 + ((long)m * 9 + i) * NCH + c4];
#pragma unroll
      for (int j = 0; j < 9; ++j) {
        float4 s = *(const float4*)&B2[base2 + (((long)m * 9 + i) * 9 + j) * NCH + c4];
        const float* b3p = &B3[base3 + ((((long)m * 9 + i) * 9 + j) * 9) * NCH + c4];
#pragma unroll
        for (int n = 0; n < 9; ++n) {
          float4 bv = *(const float4*)&b3p[(long)n * NCH];
          s.x += bv.x * xr[0][n];
          s.y += bv.y * xr[1][n];
          s.z += bv.z * xr[2][n];
          s.w += bv.w * xr[3][n];
        }
        a.x += s.x * xr[0][j];
        a.y += s.y * xr[1][j];
        a.z += s.z * xr[2][j];
        a.w += s.w * xr[3][j];
      }
      r0 += a.x * xi.x;
      r1 += a.y * xi.y;
      r2 += a.z * xi.z;
      r3 += a.w * xi.w;
    }
    float4 o = {r0, r1, r2, r3};
    *(float4*)&res[((long)m * NNODES + b) * NCH + c4] = o;
  }
}

// ---------------------------------------------------------------------------
// Phase 3: o[b,f,m] = norm * sum_c res[m][b][c] * lin[c][f]  (+ sc), using
// full-precision f32 WMMA (16x16x4, 32 k-steps for C=128).
// grid: (NNODES/16, 9)   block: 256 (8 waves, wave w -> column tile w)
// A (16x4 f32) layout: M = lane&15 for all lanes; K pair split by lane half.
// ---------------------------------------------------------------------------
__global__ void linear_wmma_kernel(const float* __restrict__ res,
                                   const float* __restrict__ lin0,
                                   const float* __restrict__ lin1,
                                   const float* __restrict__ lin2,
                                   const float* __restrict__ sc,
                                   float* __restrict__ out) {
  const int m    = blockIdx.y;
  const int r0   = blockIdx.x * 16;
  const int wave = threadIdx.x >> 5;       // 0..7 -> column tile
  const int lane = threadIdx.x & 31;
  const int half = lane >> 4;              // K sub-pair select
  const int l16  = lane & 15;
  const int f0   = wave * 16;

  const float* lin;
  int colbase, dL, dloc;
  if (m == 0)        { lin = lin0; colbase = 0;   dL = 1; dloc = 0;     }
  else if (m < 4)    { lin = lin1; colbase = 128; dL = 3; dloc = m - 1; }
  else               { lin = lin2; colbase = 512; dL = 5; dloc = m - 4; }

  const float* Arow = res + ((long)m * NNODES + r0 + l16) * NCH;

  v8f acc = {};
#pragma unroll
  for (int kb = 0; kb < 32; ++kb) {
    const int k0 = kb * 4 + half * 2;
    v2f a, bm;
    a.x  = Arow[k0];
    a.y  = Arow[k0 + 1];
    bm.x = lin[(k0) * NCH + f0 + l16];
    bm.y = lin[(k0 + 1) * NCH + f0 + l16];
    acc = __builtin_amdgcn_wmma_f32_16x16x4_f32(
        /*neg_a=*/false, a, /*neg_b=*/false, bm,
        /*c_mod=*/(short)0, acc, /*reuse_a=*/false, /*reuse_b=*/false);
  }

  const float norm = 0.08838834764831845f;   // 1/sqrt(128)
#pragma unroll
  for (int r = 0; r < 8; ++r) {
    const int row = r0 + r + half * 8;
    const int f   = f0 + l16;
    const int col = colbase + f * dL + dloc;
    const long o  = (long)row * OUTW + col;
    out[o] = acc[r] * norm + sc[o];
  }
}

// ---------------------------------------------------------------------------
extern "C" void kernel_launch(void* const* d_in, const int* in_sizes, int n_in,
                              void* d_out, int out_size, void* d_ws, size_t ws_size,
                              hipStream_t stream) {
  const float* nf    = (const float*)d_in[0];
  const float* sc    = (const float*)d_in[1];
  const float* attrs = (const float*)d_in[2];

  // per-L inputs at 3 + 7*L : U3, U2, U1, W3, W2, W1, lin
  const float* U3[3]; const float* U2[3]; const float* U1[3];
  const float* W3[3]; const float* W2[3]; const float* W1[3];
  const float* LIN[3];
  for (int L = 0; L < 3; ++L) {
    const int base = 3 + 7 * L;
    U3[L]  = (const float*)d_in[base + 0];
    U2[L]  = (const float*)d_in[base + 1];
    U1[L]  = (const float*)d_in[base + 2];
    W3[L]  = (const float*)d_in[base + 3];
    W2[L]  = (const float*)d_in[base + 4];
    W1[L]  = (const float*)d_in[base + 5];
    LIN[L] = (const float*)d_in[base + 6];
  }

  float* ws  = (float*)d_ws;
  float* B3  = ws;
  float* B2  = B3 + SZ_B3;
  float* B1  = B2 + SZ_B2;
  float* res = B1 + SZ_B1;

  // Phase 1: fused per-element basis tensors (mbase/dcount per L: {0,1},{1,3},{4,5})
  const int mbase[3]  = {0, 1, 4};
  const int dcount[3] = {1, 3, 5};
  for (int L = 0; L < 3; ++L) {
    dim3 grid(dcount[L] * 81, NE);
    precompute_kernel<<<grid, 128, 0, stream>>>(U3[L], U2[L], U1[L],
                                                W3[L], W2[L], W1[L],
                                                B3, B2, B1, mbase[L]);
  }

  // Phase 2: per-node symmetric contraction (one wave per node)
  node_contract_kernel<<<NNODES, 32, 0, stream>>>(nf, attrs, B3, B2, B1, res);

  // Phase 3: channel-mixing GEMMs via f32 WMMA + norm + sc
  {
    dim3 grid(NNODES / 16, NM);
    linear_wmma_kernel<<<grid, 256, 0, stream>>>(res, LIN[0], LIN[1], LIN[2],
                                                 sc, (float*)d_out);
  }
}